// GraphSAGELayer_58480274702577
// MI455X (gfx1250) — compile-verified
//
#include <hip/hip_runtime.h>

typedef float v2f __attribute__((ext_vector_type(2)));
typedef float v4f __attribute__((ext_vector_type(4)));
typedef float v8f __attribute__((ext_vector_type(8)));
typedef unsigned int u32x4 __attribute__((ext_vector_type(4)));
typedef int i32x4 __attribute__((ext_vector_type(4)));
typedef int i32x8 __attribute__((ext_vector_type(8)));

#define N_NODES 16384
#define D_IN    256
#define D_HID   512
#define D_OUT   256
#define KSAMP   10

// LDS row stride for the staged A-tile: 512 data words + 2 pad words
// (TDM pad: 1 DWORD after every 256 DWORDs).
#define LDSA_STRIDE 514

// Deterministic per-edge score (stand-in for the reference's uniform noise;
// top-k over i.i.d. hash scores == uniform sample without replacement).
__device__ __forceinline__ unsigned mix_hash(unsigned a, unsigned b) {
  unsigned h = a * 0x9E3779B9u ^ (b + 0x7F4A7C15u);
  h ^= h >> 16; h *= 0x7FEB352Du;
  h ^= h >> 15; h *= 0x846CA68Bu;
  h ^= h >> 16;
  return h;
}

// ---------------------------------------------------------------------------
// Kernel 1: one wave32 per node. Stream the adjacency row with non-temporal
// b128 loads (1 GiB total -> don't pollute L2), per-lane top-K in LDS,
// wave-merge via 64-bit shuffle argmax, aggregate winners, write h=[x|agg].
// ---------------------------------------------------------------------------
__global__ void __launch_bounds__(256)
sage_sample_agg(const float* __restrict__ x,
                const float* __restrict__ adj,
                float* __restrict__ h) {
  __shared__ unsigned       s_sc[256 * KSAMP];
  __shared__ unsigned short s_id[256 * KSAMP];

  const int tid  = threadIdx.x;
  const int lane = tid & 31;
  const int node = blockIdx.x * 8 + (tid >> 5);

  unsigned*       my_sc = &s_sc[tid * KSAMP];
  unsigned short* my_id = &s_id[tid * KSAMP];
  int cnt = 0;

  const size_t rowbase = (size_t)node * N_NODES;
  const v4f* adj4 = (const v4f*)(adj + rowbase);   // row base is 16B aligned
  for (int t = 0; t < N_NODES / 128; ++t) {        // 128 iters, 16B/lane
    const int vidx = (t << 5) + lane;              // coalesced 512B per step
    const v4f a = __builtin_nontemporal_load(adj4 + vidx);
    const int jb = vidx << 2;
#pragma unroll
    for (int e = 0; e < 4; ++e) {
      if (a[e] > 0.0f) {
        const int j = jb + e;
        const unsigned s = mix_hash((unsigned)node, (unsigned)j);
        if (cnt < KSAMP || s > my_sc[KSAMP - 1]) {
          int p = (cnt < KSAMP) ? cnt : (KSAMP - 1);
          while (p > 0 && my_sc[p - 1] < s) {
            my_sc[p] = my_sc[p - 1];
            my_id[p] = my_id[p - 1];
            --p;
          }
          my_sc[p] = s;
          my_id[p] = (unsigned short)j;
          if (cnt < KSAMP) ++cnt;
        }
      }
    }
  }

  float acc[D_IN / 32];
#pragma unroll
  for (int t = 0; t < D_IN / 32; ++t) acc[t] = 0.0f;

  // 10 rounds of wave-wide argmax over per-lane sorted list heads.
  // Key packs (score << 15) | (idx+1); idx+1 <= 16384 fits 15 bits, 0 = invalid.
  int pos = 0, nsel = 0;
  for (int r = 0; r < KSAMP; ++r) {
    unsigned long long key = 0ull;
    if (pos < cnt)
      key = (((unsigned long long)my_sc[pos]) << 15) |
            (unsigned)(my_id[pos] + 1);
    unsigned long long gmax = key;
#pragma unroll
    for (int off = 16; off > 0; off >>= 1) {
      unsigned long long o = __shfl_xor(gmax, off, 32);
      gmax = (o > gmax) ? o : gmax;
    }
    if (gmax == 0ull) break;      // uniform across wave
    if (key == gmax) ++pos;       // winner lane advances its head
    const int nbr = (int)(gmax & 0x7FFFull) - 1;
    const float* xr = x + (size_t)nbr * D_IN;     // x is L2-resident (16 MB)
#pragma unroll
    for (int t = 0; t < D_IN / 32; ++t) acc[t] += xr[lane + (t << 5)];
    ++nsel;
  }
  const float scale = (nsel > 0) ? (1.0f / (float)nsel) : 0.0f;

  float*       hr = h + (size_t)node * D_HID;
  const float* xs = x + (size_t)node * D_IN;
#pragma unroll
  for (int t = 0; t < D_IN / 32; ++t) {
    const int c = lane + (t << 5);
    hr[c]        = xs[c];              // self features
    hr[D_IN + c] = acc[t] * scale;     // mean-aggregated neighbors
  }
}

// ---------------------------------------------------------------------------
// Kernel 2: out = relu(h @ W^T + b) via V_WMMA_F32_16X16X4_F32.
// Block = one tm (16 rows of h) x 8 tn tiles. The 16x512 f32 A-tile (32 KB)
// is staged into LDS once per block by the Tensor Data Mover, with LDS
// padding (1 DWORD per 256) so rows sit at stride 514 -> no bank conflicts.
// All 8 waves then read A-fragments from LDS; B comes from W (L2-resident).
// ---------------------------------------------------------------------------
__global__ void __launch_bounds__(256)
sage_gemm_relu(const float* __restrict__ h,
               const float* __restrict__ W,
               const float* __restrict__ bias,
               float* __restrict__ out) {
  __shared__ float lds_h[16 * LDSA_STRIDE + 8];

  const int lane = threadIdx.x & 31;
  const int wv   = threadIdx.x >> 5;              // 0..7
  const int tm   = blockIdx.x >> 1;               // 0..1023
  const int tn   = ((blockIdx.x & 1) << 3) + wv;  // 0..15

  // --- TDM: stage h[tm*16 .. tm*16+15][0..511] (contiguous 32 KB) into LDS.
  if (wv == 0) {
    const unsigned long long ga =
        (unsigned long long)(const void*)(h + (size_t)tm * 16 * D_HID);
    const unsigned lds_off = (unsigned)(size_t)(void*)&lds_h[0];

    // D# group 0: count=1 | lds_addr | global_addr(57b) | type=2
    u32x4 g0;
    g0[0] = 1u;                                       // count = 1 descriptor
    g0[1] = lds_off;                                  // bits 63:32 lds_addr
    g0[2] = (unsigned)(ga & 0xFFFFFFFFull);           // global_addr lo
    g0[3] = (unsigned)((ga >> 32) & 0x01FFFFFFull)    // global_addr hi
          | (2u << 30);                               // type = 2 (image)

    // D# group 1: data_size=4B, pad_enable, pad_interval=256 DW, pad=1 DW,
    // one line of 8192 elements (16 contiguous rows of 512).
    i32x8 g1;
    g1[0] = (int)((2u << 16)        // data_size = 2 -> 4 bytes
                | (1u << 20)        // pad_enable
                | (7u << 22)        // pad_interval = 7 -> 256 DWORDs
                | (0u << 25));      // pad_amount   = 0 -> 1 DWORD
    g1[1] = (int)((8192u & 0xFFFFu) << 16);           // tensor_dim0 lo16
    g1[2] = (int)((8192u >> 16) | (1u << 16));        // dim0 hi | tensor_dim1=1
    g1[3] = (int)((8192u & 0xFFFFu) << 16);           // tile_dim0 = 8192
    g1[4] = 0;                                        // tile_dim1/2 unused
    g1[5] = 8192;                                     // tensor_dim0_stride
    g1[6] = 0;
    g1[7] = 0;

    const i32x4 z4 = {0, 0, 0, 0};
#if __clang_major__ >= 23
    const i32x8 z8 = {0, 0, 0, 0, 0, 0, 0, 0};
    __builtin_amdgcn_tensor_load_to_lds(g0, g1, z4, z4, z8, 0);
#else
    __builtin_amdgcn_tensor_load_to_lds(g0, g1, z4, z4, 0);
#endif
    __builtin_amdgcn_s_wait_tensorcnt(0);             // issuing wave waits
  }
  __syncthreads();                                    // release consumers

  // --- WMMA accumulation over K=512 (128 x K=4 steps).
  // A (16x4 f32): lanes 0-15 hold M rows, VGPR pair = K{0,1} / K{2,3}.
  // B (4x16 f32): lane&15 = N col, VGPR pair = K{0,1} / K{2,3}.
  const int arow  = lane & 15;
  const int ncol  = tn * 16 + (lane & 15);
  const int khalf = (lane >> 4) * 2;

  const float* wrow   = W + (size_t)ncol * D_HID;   // W row-major == W^T K-major
  const float* lds_ra = &lds_h[arow * LDSA_STRIDE];

  v8f c = {};
  for (int k = 0; k < D_HID; k += 4) {
    const int kk = k + khalf;
    v2f a   = *(const v2f*)(lds_ra + kk + (kk >> 8));   // pad-adjusted
    v2f bfr = *(const v2f*)(wrow + kk);
    // 8 args: (neg_a, A, neg_b, B, c_mod, C, reuse_a, reuse_b)
    c = __builtin_amdgcn_wmma_f32_16x16x4_f32(false, a, false, bfr,
                                              (short)0, c, false, false);
  }

  // --- Epilogue: bias + relu, C layout VGPR r <-> M=r (lanes<16) / M=r+8.
  const float bn    = bias[ncol];
  const int   mbase = tm * 16 + ((lane >> 4) << 3);
#pragma unroll
  for (int r = 0; r < 8; ++r) {
    float v = c[r] + bn;
    v = v > 0.0f ? v : 0.0f;
    __builtin_nontemporal_store(v, &out[(size_t)(mbase + r) * D_OUT + ncol]);
  }
}

extern "C" void kernel_launch(void* const* d_in, const int* in_sizes, int n_in,
                              void* d_out, int out_size, void* d_ws, size_t ws_size,
                              hipStream_t stream) {
  (void)in_sizes; (void)n_in; (void)out_size; (void)ws_size;
  const float* x   = (const float*)d_in[0];  // [16384, 256]
  const float* adj = (const float*)d_in[1];  // [16384, 16384]
  const float* W   = (const float*)d_in[2];  // [256, 512]
  const float* b   = (const float*)d_in[3];  // [256]
  // d_in[4] = sample_size (10), compile-time constant here.
  float* out = (float*)d_out;                // [16384, 256]
  float* h   = (float*)d_ws;                 // [16384, 512] = 32 MB scratch

  // Kernel 1: 8 waves/block, one wave per node.
  sage_sample_agg<<<N_NODES / 8, 256, 0, stream>>>(x, adj, h);

  // Kernel 2: block = (tm, half of tn range); 1024 * 2 blocks, 8 waves each.
  sage_gemm_relu<<<(N_NODES / 16) * 2, 256, 0, stream>>>(h, W, b, out);
}